// InversePreEmphasis_41970420417712
// MI455X (gfx1250) — compile-verified
//
#include <hip/hip_runtime.h>
#include <hip/hip_bf16.h>
#include <stdint.h>

// Problem constants (reference: B=128, C=1, T=262144, coef=0.97)
#define T_LEN   262144
#define B_ROWS  128
#define CHUNK   2048      // outputs per block
#define WARM    1024      // warm-up steps; 0.97^1024 ~ 3e-14 << fp32 eps
#define TILE    256       // floats per row per TDM tile (matches pad_interval=256 DW)
#define ROWPAD  4         // TDM pad_amount = 4 DWORDs -> row stride 260 floats (16B aligned)
#define RSTRIDE (TILE + ROWPAD)

typedef uint32_t v4u __attribute__((ext_vector_type(4)));
typedef int      v8i __attribute__((ext_vector_type(8)));
typedef int      v4i __attribute__((ext_vector_type(4)));

__device__ __forceinline__ float tanh_hw(float x) {
#if __has_builtin(__builtin_amdgcn_tanhf)
  return __builtin_amdgcn_tanhf(x);
#else
  float r;
  // gfx1250 transcendental; guard TRANS RAW hazard explicitly since the
  // compiler cannot see through the asm.
  asm volatile("v_tanh_f32 %0, %1\n\ts_delay_alu instid0(TRANS32_DEP_1)"
               : "=v"(r) : "v"(x));
  return r;
#endif
}

// Issue one TDM 2D tile load: 128 rows x TILE floats, row stride T_LEN elems,
// into LDS at lds_byte_addr with +4-DWORD padding every 256 DWORDs.
__device__ __forceinline__ void tdm_load_tile(uint32_t lds_byte_addr,
                                              const float* tile_row0) {
  uint64_t ga = (uint64_t)(uintptr_t)tile_row0;
  v4u g0;
  g0[0] = 1u;                                           // count=1, load, no gather
  g0[1] = lds_byte_addr;                                // LDS dest (bytes)
  g0[2] = (uint32_t)(ga & 0xFFFFFFFFu);                 // global_addr[31:0]
  g0[3] = (uint32_t)((ga >> 32) & 0x01FFFFFFu)          // global_addr[56:32]
        | 0x80000000u;                                  // type=2 ("image")
  v8i g1;
  g1[0] = (int)((2u << 16)      // data_size = 4 bytes
              | (1u << 20)      // pad_enable
              | (7u << 22)      // pad_interval: 256 DWORDs
              | (3u << 25));    // pad_amount: 4 DWORDs
  g1[1] = 0;                                            // tensor_dim0[15:0] = 0 (262144)
  g1[2] = (int)((T_LEN >> 16) | (B_ROWS << 16));        // tensor_dim0 hi, tensor_dim1 lo
  g1[3] = (int)(TILE << 16);                            // tensor_dim1 hi=0, tile_dim0
  g1[4] = B_ROWS;                                       // tile_dim1=128, tile_dim2=0
  g1[5] = T_LEN;                                        // tensor_dim0_stride lo32 (elems)
  g1[6] = 0;
  g1[7] = 0;
  v4i z4 = {0, 0, 0, 0};                                // 2D tensor: groups 2/3 unused
  v8i z8 = {0, 0, 0, 0, 0, 0, 0, 0};
  // clang-23 / therock-10.0 toolchain: 6-arg form
  // (v4u g0, v8i g1, v4i, v4i, v8i, i32 cpol)
  __builtin_amdgcn_tensor_load_to_lds(g0, g1, z4, z4, z8, 0);
}

__global__ void __launch_bounds__(32)
tanh_scan_chunked_kernel(const float* __restrict__ x, float* __restrict__ out) {
  __shared__ __align__(16) float lds[2][B_ROWS * RSTRIDE];  // ~260 KB of 320 KB

  const int c       = blockIdx.x;             // chunk index
  const int lane    = threadIdx.x;            // single wave32 per block
  const int t_out0  = c * CHUNK;
  const int t_start = (c == 0) ? 0 : (t_out0 - WARM);
  const int ntiles  = ((t_out0 + CHUNK) - t_start) / TILE;   // 8 or 12

  const uint32_t lds0 = (uint32_t)(uintptr_t)&lds[0][0];
  const uint32_t lds1 = (uint32_t)(uintptr_t)&lds[1][0];

  // Prologue: double-buffer two tiles in flight (ntiles >= 8 always).
  tdm_load_tile(lds0, x + (size_t)t_start);
  tdm_load_tile(lds1, x + (size_t)t_start + TILE);

  // 4 independent recurrence chains per lane (ILP to hide fma->tanh latency).
  const int r0 = lane, r1 = lane + 32, r2 = lane + 64, r3 = lane + 96;
  float h0 = 0.f, h1 = 0.f, h2 = 0.f, h3 = 0.f;

  for (int it = 0; it < ntiles; ++it) {
    if (it + 1 < ntiles) __builtin_amdgcn_s_wait_tensorcnt(1);  // tile `it` landed
    else                 __builtin_amdgcn_s_wait_tensorcnt(0);  // last tile
    asm volatile("" ::: "memory");   // keep LDS reads below the wait

    const float* buf = lds[it & 1];
    const float* bp0 = buf + r0 * RSTRIDE;
    const float* bp1 = buf + r1 * RSTRIDE;
    const float* bp2 = buf + r2 * RSTRIDE;
    const float* bp3 = buf + r3 * RSTRIDE;

    const int  tglob = t_start + it * TILE;
    const bool emit  = (tglob >= t_out0);     // warm-up tiles produce no output
    float* o0 = out + (size_t)r0 * T_LEN + tglob;
    float* o1 = out + (size_t)r1 * T_LEN + tglob;
    float* o2 = out + (size_t)r2 * T_LEN + tglob;
    float* o3 = out + (size_t)r3 * T_LEN + tglob;

#pragma unroll 4
    for (int j = 0; j < TILE; j += 4) {
      float4 a0 = *(const float4*)(bp0 + j);
      float4 a1 = *(const float4*)(bp1 + j);
      float4 a2 = *(const float4*)(bp2 + j);
      float4 a3 = *(const float4*)(bp3 + j);
      float4 y0, y1, y2, y3;
      // Row-interleaved steps: 4 independent fma->tanh chains in flight.
#define STEP(comp)                                                   \
      h0 = tanh_hw(__builtin_fmaf(0.97f, h0, a0.comp)); y0.comp = h0; \
      h1 = tanh_hw(__builtin_fmaf(0.97f, h1, a1.comp)); y1.comp = h1; \
      h2 = tanh_hw(__builtin_fmaf(0.97f, h2, a2.comp)); y2.comp = h2; \
      h3 = tanh_hw(__builtin_fmaf(0.97f, h3, a3.comp)); y3.comp = h3;
      STEP(x) STEP(y) STEP(z) STEP(w)
#undef STEP
      if (emit) {
        *(float4*)(o0 + j) = y0;
        *(float4*)(o1 + j) = y1;
        *(float4*)(o2 + j) = y2;
        *(float4*)(o3 + j) = y3;
      }
    }

    // Refill the buffer we just finished reading (single wave: program order
    // plus the value-dependencies above guarantee DS reads completed).
    if (it + 2 < ntiles) {
      asm volatile("" ::: "memory");
      tdm_load_tile((it & 1) ? lds1 : lds0,
                    x + (size_t)(tglob + 2 * TILE));
    }
  }
}

extern "C" void kernel_launch(void* const* d_in, const int* in_sizes, int n_in,
                              void* d_out, int out_size, void* d_ws, size_t ws_size,
                              hipStream_t stream) {
  const float* x = (const float*)d_in[0];   // (128, 1, 262144) fp32
  float* out = (float*)d_out;               // (128, 1, 262144) fp32
  (void)in_sizes; (void)n_in; (void)out_size; (void)d_ws; (void)ws_size;
  dim3 grid(T_LEN / CHUNK);                 // 128 chunks
  dim3 block(32);                           // one wave per block
  tanh_scan_chunked_kernel<<<grid, block, 0, stream>>>(x, out);
}